// PowerIterationLayer_41644002902558
// MI455X (gfx1250) — compile-verified
//
#include <hip/hip_runtime.h>

// ---------------------------------------------------------------------------
// Factorized power iteration (B=32, N1=N2=128, M1=M2=512, 15 iters) on
// MI455X / gfx1250 using exact-precision V_WMMA_F32_16X16X4_F32.
//
// GEMM core: 128x64 tile / WG, 8 waves, each wave owns a 2x2 grid of 16x16
// subtiles (4 accumulators) -> 1 LDS b64 fragment load per WMMA.
// Tiles are filled with CDNA5 async memory->LDS DMA
// (GLOBAL_LOAD_ASYNC_TO_LDS_B128/B32, tracked by ASYNCcnt): no VGPR staging,
// no ds_store issue slots, next chunk's fill overlaps the WMMA stream.
// Double-buffered LDS, one s_wait_asynccnt + barrier per 32-wide K chunk.
// ---------------------------------------------------------------------------

typedef float v2f __attribute__((ext_vector_type(2)));
typedef float v8f __attribute__((ext_vector_type(8)));

#define NB    32      // batch
#define DN    128     // N1 == N2
#define DM    512     // M1 == M2
#define ITERS 15
#define EPS2  1e-12f  // (1e-6)^2 : compare sum-of-squares against eps^2

enum { EPI_NONE = 0, EPI_MUL = 1, EPI_ACC_HALF = 2 };

#define WMMA_F32(a, b, c) \
  __builtin_amdgcn_wmma_f32_16x16x4_f32(false, (a), false, (b), (short)0, (c), false, false)

// async memory->LDS DMA: vdst = 32-bit LDS byte address, vaddr = 64-bit global
__device__ __forceinline__ void async_ld_b128(unsigned lds, const float* g) {
  asm volatile("global_load_async_to_lds_b128 %0, %1, off"
               :: "v"(lds), "v"(g) : "memory");
}
__device__ __forceinline__ void async_ld_b32(unsigned lds, const float* g) {
  asm volatile("global_load_async_to_lds_b32 %0, %1, off"
               :: "v"(lds), "v"(g) : "memory");
}
__device__ __forceinline__ void wait_async0() {
  asm volatile("s_wait_asynccnt 0x0" ::: "memory");
}
// generic LDS pointer -> wave-relative 32-bit LDS byte address (addr[31:0])
#define LDS_ADDR(p) ((unsigned)(size_t)(p))

// ---------------------------------------------------------------------------
//   TA: A stored [K x M] (use A[k*lda+m]);  else A stored [M x K]
//   TB: B stored [N x K] (use B[n*ldb+k]);  else B stored [K x N]
//   EPI_NONE:     C = acc
//   EPI_MUL:      C = acc * E           (fused  y = Mq ⊙ (x1·Q2))
//   EPI_ACC_HALF: C += 0.5f * acc       (fused  t += 0.5 * z1·P2ᵀ)
// ---------------------------------------------------------------------------
template <bool TA, bool TB, int EPI>
__global__ __launch_bounds__(256) void gemm_wmma_f32(
    const float* __restrict__ A, const float* __restrict__ B,
    float* __restrict__ C, const float* __restrict__ E,
    int K, int lda, int ldb, int ldc,
    long sA, long sB, long sC, long sE)
{
  // double-buffered K-major tiles; [kb][m][4] => one aligned v2f per fragment,
  // and the [4] group is 16 contiguous bytes => B128 async-DMA compatible
  __shared__ float As[2][8][128][4];   // 32 KB
  __shared__ float Bs[2][8][64][4];    // 16 KB

  const int bz = blockIdx.z;
  A += (long)bz * sA;
  B += (long)bz * sB;
  C += (long)bz * sC;
  if constexpr (EPI == EPI_MUL) E += (long)bz * sE;

  const int bm   = blockIdx.x * 128;
  const int bn   = blockIdx.y * 64;
  const int tid  = threadIdx.x;
  const int lane = tid & 31;
  const int wave = tid >> 5;

  // 2x2 subtiles per wave over the 8(M) x 4(N) subtile grid
  const int wm  = wave >> 1;            // 0..3
  const int wn  = wave & 1;             // 0..1
  const int tmA = wm * 16;              // rows  0/16/32/48
  const int tmB = tmA + 64;             // rows 64/80/96/112
  const int tnA = wn * 32;              // cols  0/32
  const int tnB = tnA + 16;             // cols 16/48
  const int l15   = lane & 15;
  const int lhalf = lane >> 4;          // 0: K 0-1 / M 0-7 ; 1: K 2-3 / M 8-15
  const int koff  = lhalf * 2;

  v8f acc00 = {0.f,0.f,0.f,0.f,0.f,0.f,0.f,0.f};
  v8f acc01 = acc00, acc10 = acc00, acc11 = acc00;

  // uniform block pointers advanced by a scalar stride per chunk
  const float* Ab = TA ? (A + bm) : (A + (long)bm * lda);
  const float* Bb = TB ? (B + (long)bn * ldb) : (B + bn);
  const long stepA = TA ? 32L * lda : 32L;
  const long stepB = TB ? 32L : 32L * ldb;

  // ---- async tile fills (A: 128x32 = 4096 f32, B: 64x32 = 2048 f32) -------
  auto fillA = [&](int buf, const float* p) {
    if constexpr (TA) {
      // [K x M]: memory contiguous along m, LDS k-interleaved -> B32/element
      #pragma unroll
      for (int j = 0; j < 16; ++j) {
        const int k = (tid >> 7) + 2 * j, m = tid & 127;
        async_ld_b32(LDS_ADDR(&As[buf][k >> 2][m][k & 3]),
                     p + (long)k * lda + m);
      }
    } else {
      // [M x K]: 4 consecutive k's are 16 contiguous bytes -> B128/quad
      #pragma unroll
      for (int j = 0; j < 4; ++j) {
        const int q = tid + j * 256;            // 0..1023 quads
        const int kb = q & 7, m = q >> 3;
        async_ld_b128(LDS_ADDR(&As[buf][kb][m][0]),
                      p + (long)m * lda + kb * 4);
      }
    }
  };
  auto fillB = [&](int buf, const float* p) {
    if constexpr (TB) {
      // [N x K]: k-contiguous -> B128/quad
      #pragma unroll
      for (int j = 0; j < 2; ++j) {
        const int q = tid + j * 256;            // 0..511 quads
        const int kb = q & 7, n = q >> 3;
        async_ld_b128(LDS_ADDR(&Bs[buf][kb][n][0]),
                      p + (long)n * ldb + kb * 4);
      }
    } else {
      // [K x N]: memory contiguous along n -> B32/element
      #pragma unroll
      for (int j = 0; j < 8; ++j) {
        const int k = (tid >> 6) + 4 * j, n = tid & 63;
        async_ld_b32(LDS_ADDR(&Bs[buf][k >> 2][n][k & 3]),
                     p + (long)k * ldb + n);
      }
    }
  };

  auto compute = [&](int buf) {
    #pragma unroll
    for (int kb = 0; kb < 8; ++kb) {
      // ISA 16x4 f32 layout: lanes 0-15 hold K=0,1; lanes 16-31 hold K=2,3
      v2f aA = *(const v2f*)&As[buf][kb][tmA + l15][koff];
      v2f aB = *(const v2f*)&As[buf][kb][tmB + l15][koff];
      v2f bA = *(const v2f*)&Bs[buf][kb][tnA + l15][koff];
      v2f bB = *(const v2f*)&Bs[buf][kb][tnB + l15][koff];
      acc00 = WMMA_F32(aA, bA, acc00);
      acc01 = WMMA_F32(aA, bB, acc01);
      acc10 = WMMA_F32(aB, bA, acc10);
      acc11 = WMMA_F32(aB, bB, acc11);
    }
  };

  // ---- pipeline: async prologue fill of buffer 0 ---------------------------
  fillA(0, Ab);
  fillB(0, Bb);
  wait_async0();
  __syncthreads();

  const int nC = K >> 5;
  for (int c = 0; c < nC; ++c) {
    const int buf = c & 1;
    const bool hasNext = (c + 1 < nC);
    if (hasNext) {           // kick next chunk's DMA before the WMMA stream
      Ab += stepA;
      Bb += stepB;
      fillA(buf ^ 1, Ab);
      fillB(buf ^ 1, Bb);
    }
    compute(buf);
    wait_async0();           // own wave's DMAs done ...
    __syncthreads();         // ... and everyone else's
  }

  // C/D layout: VGPR r: lanes 0-15 -> M=r, lanes 16-31 -> M=8+r; N = lane&15
  const int colA = bn + tnA + l15;
  const int colB = bn + tnB + l15;
  #pragma unroll
  for (int r = 0; r < 8; ++r) {
    const int rowA = bm + tmA + lhalf * 8 + r;
    const int rowB = bm + tmB + lhalf * 8 + r;
    const long oAA = (long)rowA * ldc + colA;
    const long oAB = (long)rowA * ldc + colB;
    const long oBA = (long)rowB * ldc + colA;
    const long oBB = (long)rowB * ldc + colB;
    if constexpr (EPI == EPI_NONE) {
      C[oAA] = acc00[r]; C[oAB] = acc01[r];
      C[oBA] = acc10[r]; C[oBB] = acc11[r];
    } else if constexpr (EPI == EPI_MUL) {
      C[oAA] = acc00[r] * E[oAA]; C[oAB] = acc01[r] * E[oAB];
      C[oBA] = acc10[r] * E[oBA]; C[oBB] = acc11[r] * E[oBB];
    } else {
      C[oAA] += 0.5f * acc00[r]; C[oAB] += 0.5f * acc01[r];
      C[oBA] += 0.5f * acc10[r]; C[oBB] += 0.5f * acc11[r];
    }
  }
}

// ---------------------------------------------------------------------------
// Elementwise / reduction kernels
// ---------------------------------------------------------------------------
__global__ __launch_bounds__(256) void init_kernel(float4* v, int* active, int n4) {
  const int i = blockIdx.x * blockDim.x + threadIdx.x;
  if (i < n4) v[i] = make_float4(1.f, 1.f, 1.f, 1.f);
  if (i == 0) *active = 1;
}

__global__ __launch_bounds__(256) void tinit_kernel(const float4* __restrict__ Mp,
                                                    const float4* __restrict__ v,
                                                    float4* __restrict__ t, int n4) {
  const int i = blockIdx.x * blockDim.x + threadIdx.x;
  if (i < n4) {
    const float4 a = Mp[i], b = v[i];
    t[i] = make_float4(a.x * b.x, a.y * b.y, a.z * b.z, a.w * b.w);
  }
}

// one block per batch: nrm[b] = sum(t^2) over 128*128 elems (deterministic)
__global__ __launch_bounds__(256) void nrm_kernel(const float* __restrict__ t,
                                                  float* __restrict__ nrm) {
  __shared__ float red[256];
  const long base = (long)blockIdx.x * (DN * DN);
  float s = 0.f;
  #pragma unroll 4
  for (int i = 0; i < (DN * DN) / 256; ++i) {
    const float x = t[base + threadIdx.x + i * 256];
    s += x * x;
  }
  red[threadIdx.x] = s;
  __syncthreads();
  for (int off = 128; off > 0; off >>= 1) {
    if (threadIdx.x < off) red[threadIdx.x] += red[threadIdx.x + off];
    __syncthreads();
  }
  if (threadIdx.x == 0) nrm[blockIdx.x] = red[0];
}

// one block per batch: u = div_no_nan(t, sqrt(nrm)); diff[b] = ||u - v||^2;
// v = active ? u : v   (gating uses the flag coming INTO this step)
__global__ __launch_bounds__(256) void update_kernel(const float* __restrict__ t,
                                                     float* __restrict__ v,
                                                     const float* __restrict__ nrm,
                                                     float* __restrict__ diff,
                                                     const int* __restrict__ active) {
  __shared__ float red[256];
  const int b = blockIdx.x;
  const float nn  = nrm[b];
  const float inv = (nn == 0.f) ? 0.f : (1.0f / sqrtf(nn));
  const int act = *active;
  const long base = (long)b * (DN * DN);
  float s = 0.f;
  #pragma unroll 4
  for (int i = 0; i < (DN * DN) / 256; ++i) {
    const long idx = base + threadIdx.x + i * 256;
    const float u  = t[idx] * inv;
    const float vo = v[idx];
    const float d  = u - vo;
    s += d * d;
    v[idx] = act ? u : vo;
  }
  red[threadIdx.x] = s;
  __syncthreads();
  for (int off = 128; off > 0; off >>= 1) {
    if (threadIdx.x < off) red[threadIdx.x] += red[threadIdx.x + off];
    __syncthreads();
  }
  if (threadIdx.x == 0) diff[b] = red[0];
}

// active = active && any_b(diff[b] >= eps^2)
__global__ void flag_kernel(const float* __restrict__ diff, int* active) {
  __shared__ int anyf;
  if (threadIdx.x == 0) anyf = 0;
  __syncthreads();
  if (threadIdx.x < NB && diff[threadIdx.x] >= EPS2) atomicOr(&anyf, 1);
  __syncthreads();
  if (threadIdx.x == 0) *active = (*active) && anyf;
}

// ---------------------------------------------------------------------------
// Host driver
// ---------------------------------------------------------------------------
extern "C" void kernel_launch(void* const* d_in, const int* in_sizes, int n_in,
                              void* d_out, int out_size, void* d_ws, size_t ws_size,
                              hipStream_t stream) {
  (void)in_sizes; (void)n_in; (void)out_size; (void)ws_size;

  const float* Mp = (const float*)d_in[0];   // [B,128,128]
  const float* Mq = (const float*)d_in[1];   // [B,512,512]
  const float* G1 = (const float*)d_in[2];   // [B,128,512]
  const float* G2 = (const float*)d_in[3];   // [B,128,512]
  const float* H1 = (const float*)d_in[4];   // [B,128,512]
  const float* H2 = (const float*)d_in[5];   // [B,128,512]

  // workspace layout (bytes)
  char* ws = (char*)d_ws;
  float* v    = (float*)(ws + 0x0000000);  //  2 MB  [B,128,128]
  float* t    = (float*)(ws + 0x0200000);  //  2 MB  [B,128,128]
  float* x1   = (float*)(ws + 0x0400000);  //  8 MB  [B,512,128]
  float* y    = (float*)(ws + 0x0C00000);  // 32 MB  [B,512,512]
  float* z1   = (float*)(ws + 0x2C00000);  //  8 MB  [B,128,512]
  float* nrm  = (float*)(ws + 0x3400000);  // [B]
  float* diff = nrm + NB;                  // [B]
  int*   act  = (int*)(diff + NB);

  const int  n4  = NB * DN * DN / 4;       // 131072 float4
  const long sV  = (long)DN * DN;          // 16384
  const long sGH = (long)DN * DM;          // 65536
  const long sMq = (long)DM * DM;          // 262144

  const dim3 blk(256);
  const dim3 gEw((n4 + 255) / 256);
  const dim3 g1(DM / 128, DN / 64, NB);    // x1 : 512x128 -> (4,2,32)
  const dim3 g2(DM / 128, DM / 64, NB);    // y  : 512x512 -> (4,8,32)
  const dim3 g3(DN / 128, DM / 64, NB);    // z1 : 128x512 -> (1,8,32)
  const dim3 g4(DN / 128, DN / 64, NB);    // t  : 128x128 -> (1,2,32)

  init_kernel<<<gEw, blk, 0, stream>>>((float4*)v, act, n4);

  for (int it = 0; it < ITERS; ++it) {
    // t = Mp ⊙ v
    tinit_kernel<<<gEw, blk, 0, stream>>>((const float4*)Mp, (const float4*)v,
                                          (float4*)t, n4);

    // two mtr passes: (Q1,Q2,P1,P2) = (H1,H2,G1,G2) then (G1,G2,H1,H2)
    const float* Q1[2] = {H1, G1};
    const float* Q2[2] = {H2, G2};
    const float* P1[2] = {G1, H1};
    const float* P2[2] = {G2, H2};
    for (int p = 0; p < 2; ++p) {
      // x1 = Q1ᵀ v            [512,128] = [128,512]ᵀ · [128,128]
      gemm_wmma_f32<true, false, EPI_NONE><<<g1, blk, 0, stream>>>(
          Q1[p], v, x1, nullptr, DN, DM, DN, DN, sGH, sV, sGH, 0);
      // y = Mq ⊙ (x1 · Q2)    [512,512] = [512,128] · [128,512]
      gemm_wmma_f32<false, false, EPI_MUL><<<g2, blk, 0, stream>>>(
          x1, Q2[p], y, Mq, DN, DN, DM, DM, sGH, sGH, sMq, sMq);
      // z1 = P1 · y           [128,512] = [128,512] · [512,512]
      gemm_wmma_f32<false, false, EPI_NONE><<<g3, blk, 0, stream>>>(
          P1[p], y, z1, nullptr, DM, DM, DM, DM, sGH, sMq, sGH, 0);
      // t += 0.5 · z1 · P2ᵀ   [128,128] += [128,512] · [128,512]ᵀ
      gemm_wmma_f32<false, true, EPI_ACC_HALF><<<g4, blk, 0, stream>>>(
          z1, P2[p], t, nullptr, DM, DM, DM, DN, sGH, sGH, sV, 0);
    }

    nrm_kernel<<<dim3(NB), blk, 0, stream>>>(t, nrm);
    update_kernel<<<dim3(NB), blk, 0, stream>>>(t, v, nrm, diff, act);
    flag_kernel<<<dim3(1), dim3(32), 0, stream>>>(diff, act);
  }

  hipMemcpyAsync(d_out, v, (size_t)NB * DN * DN * sizeof(float),
                 hipMemcpyDeviceToDevice, stream);
}